// ImageReconstruction_55825984913931
// MI455X (gfx1250) — compile-verified
//
#include <hip/hip_runtime.h>

// Reference shapes
#define N_SEQ   512
#define TILES   1024
#define NBLK    512
#define IMGW    256
#define BATCH   32

typedef float v2f __attribute__((ext_vector_type(2)));
typedef float v8f __attribute__((ext_vector_type(8)));

__global__ __launch_bounds__(256)
void ImageReconstruction_kernel(const int*   __restrict__ frame_idxs,
                                const float* __restrict__ sequence,
                                const float* __restrict__ blocks,
                                const float* __restrict__ gumbel,
                                float*       __restrict__ out)
{
    __shared__ int s_idx[16];

    const int lane  = threadIdx.x & 31;
    const int wave  = threadIdx.x >> 5;
    const int group = blockIdx.x;          // 0 .. 2047 (each = 16 flat (b,l) rows)
    const int g0    = group << 4;          // first flat row of this group
    const int b     = g0 >> 10;            // TILES == 1024 rows per batch image
    const int l0    = g0 & (TILES - 1);

    const int frame = frame_idxs[b];
    const float* seqbase = sequence + (size_t)frame * (size_t)(TILES * NBLK);
    const float* gumbase = gumbel   + (size_t)b     * (size_t)(TILES * NBLK);

    // ---------------- Stage 1: argmax over 512 logits+gumbel per row ----------------
    // 8 waves x 2 rows each = 16 rows. Fully coalesced float4 streaming.
    #pragma unroll
    for (int rr = 0; rr < 2; ++rr) {
        const int r = wave * 2 + rr;
        const float* srow = seqbase + (size_t)(l0 + r) * NBLK;
        const float* grow = gumbase + (size_t)(l0 + r) * NBLK;

        float vmax = -3.402823466e38f;
        int   imax = 0;
        #pragma unroll
        for (int it = 0; it < 4; ++it) {
            const int e = it * 128 + lane * 4;
            const float4 s = *(const float4*)(srow + e);
            const float4 g = *(const float4*)(grow + e);
            float v;
            v = s.x + g.x; if (v > vmax) { vmax = v; imax = e + 0; }
            v = s.y + g.y; if (v > vmax) { vmax = v; imax = e + 1; }
            v = s.z + g.z; if (v > vmax) { vmax = v; imax = e + 2; }
            v = s.w + g.w; if (v > vmax) { vmax = v; imax = e + 3; }
        }
        // wave32 butterfly; lowest index wins ties (matches jnp.argmax)
        #pragma unroll
        for (int m = 16; m >= 1; m >>= 1) {
            const float ov = __shfl_xor(vmax, m, 32);
            const int   oi = __shfl_xor(imax, m, 32);
            if (ov > vmax || (ov == vmax && oi < imax)) { vmax = ov; imax = oi; }
        }
        if (lane == 0) s_idx[r] = imax;
    }
    __syncthreads();

    // ---------------- Stage 2: one-hot x blocks via V_WMMA_F32_16X16X4_F32 ----------------
    // wave 0 computes the 16x64 tile exactly (one-hot A => exact f32 arithmetic)
    if (wave == 0) {
        const int nloc = lane & 15;        // M for A, N for B/C/D
        const int hi   = lane >> 4;        // half-wave selector
        const int k0   = hi ? 2 : 0;       // K values held by this half (A & B layout)

        v8f acc0 = {}, acc1 = {}, acc2 = {}, acc3 = {};

        for (int t = 0; t < 16; ++t) {
            const int idx   = s_idx[t];
            const int chunk = idx >> 2;    // which group of 4 K-values
            const int pos   = idx & 3;     // position inside the group

            // A (16x4): single 1.0 at (M=t, K=pos)
            v2f a; a.x = 0.0f; a.y = 0.0f;
            if (nloc == t) {
                a.x = (pos == k0    ) ? 1.0f : 0.0f;
                a.y = (pos == k0 + 1) ? 1.0f : 0.0f;
            }

            // B (4x16 slab of blocks[chunk*4 .. chunk*4+3][cc*16 .. cc*16+15])
            const float* bb = blocks + (size_t)(chunk * 4 + k0) * 64;
            v2f b0, b1, b2, b3;
            b0.x = bb[ 0 + nloc]; b0.y = bb[64 +  0 + nloc];
            b1.x = bb[16 + nloc]; b1.y = bb[64 + 16 + nloc];
            b2.x = bb[32 + nloc]; b2.y = bb[64 + 32 + nloc];
            b3.x = bb[48 + nloc]; b3.y = bb[64 + 48 + nloc];

            acc0 = __builtin_amdgcn_wmma_f32_16x16x4_f32(false, a, false, b0, (short)0, acc0, false, false);
            acc1 = __builtin_amdgcn_wmma_f32_16x16x4_f32(false, a, false, b1, (short)0, acc1, false, false);
            acc2 = __builtin_amdgcn_wmma_f32_16x16x4_f32(false, a, false, b2, (short)0, acc2, false, false);
            acc3 = __builtin_amdgcn_wmma_f32_16x16x4_f32(false, a, false, b3, (short)0, acc3, false, false);
        }

        // ---------------- Scatter 16x64 tile into the 256x256 image ----------------
        // C/D layout: VGPR v, lanes 0-15 -> M=v, lanes 16-31 -> M=v+8; N = lane&15
        #pragma unroll
        for (int v = 0; v < 8; ++v) {
            const int M  = hi ? (v + 8) : v;
            const int l  = l0 + M;
            const int ty = l >> 5;          // 32 tiles per image row
            const int tx = l & 31;
            const float vals[4] = { acc0[v], acc1[v], acc2[v], acc3[v] };
            #pragma unroll
            for (int cc = 0; cc < 4; ++cc) {
                const int colu = cc * 16 + nloc;     // 0..63 within the 8x8 block
                const int i = colu >> 3;
                const int j = colu & 7;
                out[((size_t)b * IMGW + (size_t)(ty * 8 + i)) * IMGW + (tx * 8 + j)] = vals[cc];
            }
        }
    }
}

extern "C" void kernel_launch(void* const* d_in, const int* in_sizes, int n_in,
                              void* d_out, int out_size, void* d_ws, size_t ws_size,
                              hipStream_t stream) {
    const int*   frame_idxs = (const int*)  d_in[0];
    const float* sequence   = (const float*)d_in[1];
    const float* blocks     = (const float*)d_in[2];
    const float* gumbel     = (const float*)d_in[3];
    float*       out        = (float*)d_out;

    const int groups = (BATCH * TILES) / 16;   // 2048
    ImageReconstruction_kernel<<<groups, 256, 0, stream>>>(frame_idxs, sequence, blocks, gumbel, out);
}